// MultiHeadAttention_55662776156607
// MI455X (gfx1250) — compile-verified
//
#include <hip/hip_runtime.h>
#include <hip/hip_bf16.h>
#include <math.h>

#define BATCH 2
#define SEQ   2048
#define DM    1024
#define NQ    16
#define NKV   4
#define GRPQ  4
#define DH    64

typedef __attribute__((ext_vector_type(2))) float v2f;
typedef __attribute__((ext_vector_type(8))) float v8f;

__device__ __forceinline__ v8f wmma_f32(v2f a, v2f b, v8f c) {
  // V_WMMA_F32_16X16X4_F32: D = A(16x4) * B(4x16) + C, full fp32
  return __builtin_amdgcn_wmma_f32_16x16x4_f32(false, a, false, b, (short)0, c, false, false);
}

// ---------------- LayerNorm: one 256-thread block per row of 1024 ----------------
__global__ void ln_rows(const float* __restrict__ X, const float* __restrict__ gamma,
                        const float* __restrict__ beta, float* __restrict__ Y) {
  __shared__ float s1[256], s2[256];
  const int row = blockIdx.x;
  const float* x = X + (size_t)row * DM;
  float* y = Y + (size_t)row * DM;
  float a = 0.f, b = 0.f;
  for (int i = threadIdx.x; i < DM; i += 256) { float v = x[i]; a += v; b += v * v; }
  s1[threadIdx.x] = a; s2[threadIdx.x] = b;
  __syncthreads();
  for (int off = 128; off > 0; off >>= 1) {
    if (threadIdx.x < off) { s1[threadIdx.x] += s1[threadIdx.x + off];
                             s2[threadIdx.x] += s2[threadIdx.x + off]; }
    __syncthreads();
  }
  const float mu  = s1[0] * (1.0f / DM);
  const float var = s2[0] * (1.0f / DM) - mu * mu;
  const float inv = rsqrtf(var + 1e-5f);
  for (int i = threadIdx.x; i < DM; i += 256)
    y[i] = (x[i] - mu) * inv * gamma[i] + beta[i];
}

// ------- fp32 WMMA GEMM, 16x64 tile per wave: C[MxN] = A[MxK] * W[KxN] + bias -------
__global__ void gemm_f32_wmma(const float* __restrict__ A, const float* __restrict__ W,
                              const float* __restrict__ bias, float* __restrict__ C,
                              int M, int N, int K) {
  const int lane = threadIdx.x & 31;
  const int wave = threadIdx.x >> 5;
  const int tiles_n = N >> 6;                 // 64-wide N tiles
  const int tiles_m = M >> 4;
  const int tile = blockIdx.x * 8 + wave;
  if (tile >= tiles_m * tiles_n) return;
  const int mt = tile / tiles_n, nt = tile % tiles_n;
  const int half = lane >> 4, lm = lane & 15, koff = half * 2;
  const int col0 = nt * 64 + lm;
  const float* arow = A + (size_t)(mt * 16 + lm) * K;
  v8f c0 = {}, c1 = {}, c2 = {}, c3 = {};
  for (int kb = 0; kb < K; kb += 4) {
    const v2f a = *(const v2f*)(arow + kb + koff);        // global_load_b64
    const float* w0 = W + (size_t)(kb + koff)     * N + col0;
    const float* w1 = W + (size_t)(kb + koff + 1) * N + col0;
    v2f b0, b1, b2, b3;
    b0.x = w0[0];  b0.y = w1[0];
    b1.x = w0[16]; b1.y = w1[16];
    b2.x = w0[32]; b2.y = w1[32];
    b3.x = w0[48]; b3.y = w1[48];
    c0 = wmma_f32(a, b0, c0);
    c1 = wmma_f32(a, b1, c1);
    c2 = wmma_f32(a, b2, c2);
    c3 = wmma_f32(a, b3, c3);
  }
  const float bb0 = bias[col0], bb1 = bias[col0 + 16];
  const float bb2 = bias[col0 + 32], bb3 = bias[col0 + 48];
#pragma unroll
  for (int r = 0; r < 8; ++r) {
    float* crow = C + (size_t)(mt * 16 + half * 8 + r) * N + col0;
    crow[0]  = c0[r] + bb0;
    crow[16] = c1[r] + bb1;
    crow[32] = c2[r] + bb2;
    crow[48] = c3[r] + bb3;
  }
}

// ---------------- RoPE in place: X is (B,S,nheads*64), one thread per even/odd pair ----------------
__global__ void rope_inplace(float* __restrict__ X, int nheads) {
  const int idx = blockIdx.x * blockDim.x + threadIdx.x;
  const int total = BATCH * SEQ * nheads * (DH / 2);
  if (idx >= total) return;
  const int j  = idx & 31;
  const int hd = (idx >> 5) % nheads;
  const int t  = (idx / (32 * nheads)) % SEQ;
  const int bb = idx / (32 * nheads * SEQ);
  const float ang = (float)t * powf(10000.0f, -(float)(2 * j) / (float)DH);
  float sn, cs;
  sincosf(ang, &sn, &cs);
  const size_t base = (((size_t)(bb * SEQ + t) * nheads + hd) * DH) + 2 * j;
  const float x1 = X[base], x2 = X[base + 1];
  X[base]     = x1 * cs - x2 * sn;
  X[base + 1] = x1 * sn + x2 * cs;
}

// ---------------- Scores: S = Q Kᵀ / 8 for lower-triangular 16x16 tiles only ----------------
__global__ void scores_wmma(const float* __restrict__ Q, const float* __restrict__ Km,
                            float* __restrict__ attn) {
  const int lane = threadIdx.x & 31;
  const int wave = threadIdx.x >> 5;
  const int tile = blockIdx.x * 8 + wave;
  const int Ttile = tile & 127;
  int rest = tile >> 7;
  const int ttile = rest & 127;
  rest >>= 7;
  const int qh = rest & 15;
  const int bb = rest >> 4;
  if (Ttile > ttile) return;          // fully above causal diagonal: never needed
  const int h = qh >> 2, H = qh & 3;  // q head = h*GRP + H
  const int half = lane >> 4, lm = lane & 15, koff = half * 2;
  const int tb = ttile * 16, Tb = Ttile * 16;
  const float* qrow = Q  + ((size_t)(bb * SEQ + tb + lm) * NQ  + qh) * DH;
  const float* krow = Km + ((size_t)(bb * SEQ + Tb + lm) * NKV + h)  * DH;
  v8f c = {};
#pragma unroll
  for (int kb = 0; kb < DH; kb += 4) {
    const v2f a = *(const v2f*)(qrow + kb + koff);
    const v2f b = *(const v2f*)(krow + kb + koff);
    c = wmma_f32(a, b, c);
  }
  // attn_viz layout: (b, H, h, t, T)
  const size_t abase = ((size_t)((bb * GRPQ + H) * NKV + h)) * SEQ * SEQ;
  float* out = attn + abase + (size_t)tb * SEQ + Tb;
#pragma unroll
  for (int r = 0; r < 8; ++r)
    out[(size_t)(half * 8 + r) * SEQ + lm] = c[r] * 0.125f;
}

// -------- Row softmax, register-cached (1 read + 1 write per element) --------
__global__ void softmax_rows(float* __restrict__ attn, const int* __restrict__ kvlen) {
  __shared__ float red[256];
  const int rid = blockIdx.x;
  const int t  = rid % SEQ;
  const int hf = (rid / SEQ) % NQ;   // storage head index = H*NKV + h
  const int bb = rid / (SEQ * NQ);
  float* row = attn + ((size_t)(bb * NQ + hf) * SEQ + t) * SEQ;
  const int len = min(t + 1, kvlen[bb]);   // kvlen >= S/2, so len >= 1
  float vals[SEQ / 256];
  float m = -INFINITY;
#pragma unroll
  for (int j = 0; j < SEQ / 256; ++j) {
    const int i = threadIdx.x + j * 256;
    vals[j] = (i < len) ? row[i] : -INFINITY;
    m = fmaxf(m, vals[j]);
  }
  red[threadIdx.x] = m; __syncthreads();
  for (int off = 128; off > 0; off >>= 1) {
    if (threadIdx.x < off) red[threadIdx.x] = fmaxf(red[threadIdx.x], red[threadIdx.x + off]);
    __syncthreads();
  }
  m = red[0]; __syncthreads();
  float sum = 0.f;
#pragma unroll
  for (int j = 0; j < SEQ / 256; ++j) {
    vals[j] = expf(vals[j] - m);           // masked: expf(-inf) == 0 exactly
    sum += vals[j];
  }
  red[threadIdx.x] = sum; __syncthreads();
  for (int off = 128; off > 0; off >>= 1) {
    if (threadIdx.x < off) red[threadIdx.x] += red[threadIdx.x + off];
    __syncthreads();
  }
  const float inv = 1.0f / red[0];
#pragma unroll
  for (int j = 0; j < SEQ / 256; ++j)
    row[threadIdx.x + j * 256] = vals[j] * inv;
}

// -------- seq = attn @ V: 16x64 tile per wave, attn read once, K capped at causal frontier --------
__global__ void attn_v_wmma(const float* __restrict__ attn, const float* __restrict__ V,
                            float* __restrict__ seq) {
  const int lane = threadIdx.x & 31;
  const int wave = threadIdx.x >> 5;
  const int tile = blockIdx.x * 8 + wave;
  const int ttile = tile & 127;
  int rest = tile >> 7;
  const int qh = rest & 15;
  const int bb = rest >> 4;
  const int h = qh >> 2, H = qh & 3;
  const int half = lane >> 4, lm = lane & 15, koff = half * 2;
  const int tb = ttile * 16;
  const size_t abase = ((size_t)((bb * GRPQ + H) * NKV + h)) * SEQ * SEQ;
  const float* arow = attn + abase + (size_t)(tb + lm) * SEQ;
  const int kmax = tb + 16;   // attn is exactly 0 beyond the causal frontier
  v8f c0 = {}, c1 = {}, c2 = {}, c3 = {};
  for (int kb = 0; kb < kmax; kb += 4) {
    const v2f a = *(const v2f*)(arow + kb + koff);
    const float* v0 = V + ((size_t)(bb * SEQ + kb + koff)     * NKV + h) * DH + lm;
    const float* v1 = V + ((size_t)(bb * SEQ + kb + koff + 1) * NKV + h) * DH + lm;
    v2f b0, b1, b2, b3;
    b0.x = v0[0];  b0.y = v1[0];
    b1.x = v0[16]; b1.y = v1[16];
    b2.x = v0[32]; b2.y = v1[32];
    b3.x = v0[48]; b3.y = v1[48];
    c0 = wmma_f32(a, b0, c0);
    c1 = wmma_f32(a, b1, c1);
    c2 = wmma_f32(a, b2, c2);
    c3 = wmma_f32(a, b3, c3);
  }
#pragma unroll
  for (int r = 0; r < 8; ++r) {
    const int t = tb + half * 8 + r;
    float* srow = seq + ((size_t)(bb * SEQ + t) * NQ + qh) * DH + lm;
    srow[0]  = c0[r];
    srow[16] = c1[r];
    srow[32] = c2[r];
    srow[48] = c3[r];
  }
}

extern "C" void kernel_launch(void* const* d_in, const int* in_sizes, int n_in,
                              void* d_out, int out_size, void* d_ws, size_t ws_size,
                              hipStream_t stream) {
  const float* q_in  = (const float*)d_in[0];
  const float* k_in  = (const float*)d_in[1];
  const float* v_in  = (const float*)d_in[2];
  const int*   kvlen = (const int*)  d_in[3];
  const float* gamma = (const float*)d_in[4];
  const float* beta  = (const float*)d_in[5];
  const float* Wq = (const float*)d_in[6];
  const float* bq = (const float*)d_in[7];
  const float* Wk = (const float*)d_in[8];
  const float* bk = (const float*)d_in[9];
  const float* Wv = (const float*)d_in[10];
  const float* bv = (const float*)d_in[11];
  const float* Wo = (const float*)d_in[12];
  const float* bo = (const float*)d_in[13];

  const size_t ROWS = (size_t)BATCH * SEQ;          // 4096
  float* ws  = (float*)d_ws;
  float* lnq = ws;
  float* lnk = lnq + ROWS * DM;
  float* lnv = lnk + ROWS * DM;
  float* Q   = lnv + ROWS * DM;                     // (B,S,NQ*64)
  float* K   = Q   + ROWS * NQ * DH;                // (B,S,NKV*64)
  float* V   = K   + ROWS * NKV * DH;
  float* seq = V   + ROWS * NKV * DH;               // (B,S,NQ*64)

  float* out  = (float*)d_out;                      // (B,S,DM)
  float* attn = out + ROWS * DM;                    // (B,GRP,NKV,S,S)

  // 1) LayerNorm all three inputs
  ln_rows<<<BATCH * SEQ, 256, 0, stream>>>(q_in, gamma, beta, lnq);
  ln_rows<<<BATCH * SEQ, 256, 0, stream>>>(k_in, gamma, beta, lnk);
  ln_rows<<<BATCH * SEQ, 256, 0, stream>>>(v_in, gamma, beta, lnv);

  // 2) Projections (fp32 WMMA, 16x64 tiles). Tiles = (M/16)*(N/64), 8 waves/block.
  {
    int tiles = (4096 / 16) * (1024 / 64);          // Q: 4096
    gemm_f32_wmma<<<tiles / 8, 256, 0, stream>>>(lnq, Wq, bq, Q, 4096, 1024, 1024);
    tiles = (4096 / 16) * (256 / 64);               // K/V: 1024
    gemm_f32_wmma<<<tiles / 8, 256, 0, stream>>>(lnk, Wk, bk, K, 4096, 256, 1024);
    gemm_f32_wmma<<<tiles / 8, 256, 0, stream>>>(lnv, Wv, bv, V, 4096, 256, 1024);
  }

  // 3) RoPE on Q and K (in place)
  {
    int totq = BATCH * SEQ * NQ  * (DH / 2);        // 2,097,152
    int totk = BATCH * SEQ * NKV * (DH / 2);        // 524,288
    rope_inplace<<<(totq + 255) / 256, 256, 0, stream>>>(Q, NQ);
    rope_inplace<<<(totk + 255) / 256, 256, 0, stream>>>(K, NKV);
  }

  // 4) Scores (lower-triangular tiles only), scaled by 1/sqrt(64)
  {
    int tiles = BATCH * NQ * 128 * 128;             // 524,288
    scores_wmma<<<tiles / 8, 256, 0, stream>>>(Q, K, attn);
  }

  // 5) Softmax rows (applies causal + padding mask, zeros masked region)
  softmax_rows<<<BATCH * NQ * SEQ, 256, 0, stream>>>(attn, kvlen);

  // 6) seq = attn @ V  (16x64 per wave)
  {
    int tiles = BATCH * NQ * 128;                   // 4096 waves
    attn_v_wmma<<<tiles / 8, 256, 0, stream>>>(attn, V, seq);
  }

  // 7) out = seq @ Wo + bo
  {
    int tiles = (4096 / 16) * (1024 / 64);          // 4096
    gemm_f32_wmma<<<tiles / 8, 256, 0, stream>>>(seq, Wo, bo, out, 4096, 1024, 1024);
  }
}